// MultiBoxLoss_31310311588541
// MI455X (gfx1250) — compile-verified
//
#include <hip/hip_runtime.h>
#include <math.h>

// MultiBoxLoss (RetinaFace-style) for MI455X / gfx1250, wave32.
// B=16, P=65536, O=32, C=2. Memory-bound (~76MB fp32 in). Strategy:
//  - single fused streaming pass for match+encode+losses (loc/landm/dpt only
//    touched for positive priors)
//  - radix-histogram selection replaces the argsort/argsort hard-negative mining
//    (exact: tied negatives at the threshold all have ce == loss_c == t)
//  - deterministic reductions; cross-lane sums via V_WMMA_F32_16X16X4_F32
//    (D = values(16x4) * ones(4x16): row m sums lane m + lane m+16 partials),
//    finished with one ds_swizzle SWAPX16
//  - GT staging to LDS via GLOBAL_LOAD_ASYNC_TO_LDS_B32 (ASYNCcnt) when available

#define NB 16
#define NP 65536
#define NO 32
#define THRESH  0.35f
#define NEGPOS  7
#define VAR0    0.1f
#define VAR1    0.2f

typedef float v2f __attribute__((ext_vector_type(2)));
typedef float v8f __attribute__((ext_vector_type(8)));

#if defined(__AMDGCN__) && __has_builtin(__builtin_amdgcn_global_load_async_to_lds_b32)
#define HAVE_ASYNC_LDS 1
// Per-lane async global->LDS dword copy (tracked with ASYNCcnt, no VGPR data).
// Toolchain declares the builtin with generic int* params; clang inserts the
// address-space casts.
__device__ __forceinline__ void async_copy_b32(float* lds_dst, const float* gsrc) {
  __builtin_amdgcn_global_load_async_to_lds_b32(
      (int*)gsrc, (int*)lds_dst, /*offset=*/0, /*cpol=*/0);
}
__device__ __forceinline__ void wait_async0() {
#if __has_builtin(__builtin_amdgcn_s_wait_asynccnt)
  __builtin_amdgcn_s_wait_asynccnt(0);
#else
  asm volatile("s_wait_asynccnt 0x0" ::: "memory");
#endif
}
#endif

// Sum of the 32 per-lane values of a wave via one f32 WMMA.
// A-fragment layout (16x4 f32): lanes 0-15 hold (K0,K1) of row M=lane,
// lanes 16-31 hold (K2,K3) of row M=lane-16. With B=ones, D[m][*] =
// s(lane m) + s(lane m+16); summing the 8 D regs per lane gives the half-sum,
// ds_swizzle SWAPX16 (xor 0x10, and 0x1f -> imm 0x401F) completes it.
// Requires full EXEC (all call sites are convergent).
__device__ __forceinline__ float wave_sum32(float s) {
  v2f a;  a[0] = s;   a[1] = 0.f;
  v2f bo; bo[0] = 1.f; bo[1] = 1.f;
  v8f c = {};
  v8f d = __builtin_amdgcn_wmma_f32_16x16x4_f32(
      /*neg_a=*/false, a, /*neg_b=*/false, bo,
      /*c_mod=*/(short)0, c, /*reuse_a=*/false, /*reuse_b=*/false);
  float t = d[0] + d[1] + d[2] + d[3] + d[4] + d[5] + d[6] + d[7];
  t += __int_as_float(__builtin_amdgcn_ds_swizzle(__float_as_int(t), 0x401F));
  return t;
}

__device__ __forceinline__ float smoothl1(float x) {
  float d = fabsf(x);
  return d < 1.f ? 0.5f * d * d : d - 0.5f;
}

// ---------------------------------------------------------------- init
__global__ void k_init(unsigned* hist, int* npos, int* done) {
  int tid = threadIdx.x;
  for (int i = tid; i < NB * 256; i += 256) hist[i] = 0u;
  if (tid < NB) { npos[tid] = 0; done[tid] = 0; }
}

// ------------------------------------------------- per-GT best prior (argmax over P)
__global__ void k_bestprior(const float* __restrict__ priors,
                            const float* __restrict__ boxes_gt,
                            int* __restrict__ bpidx, int* __restrict__ bpvalid) {
  __shared__ float sv[256];
  __shared__ int   si[256];
  int blk = blockIdx.x;            // b*NO + o
  int tid = threadIdx.x;
  const float* g = boxes_gt + blk * 4;
  float gx1 = g[0], gy1 = g[1], gx2 = g[2], gy2 = g[3];
  float aa = (gx2 - gx1) * (gy2 - gy1);
  float best = -1.f; int bi = 0x7FFFFFFF;
  for (int p = tid; p < NP; p += 256) {
    if (p + 4096 < NP) __builtin_prefetch(priors + 4 * (p + 4096), 0, 1);
    float4 pr = reinterpret_cast<const float4*>(priors)[p];
    float px1 = pr.x - 0.5f * pr.z, py1 = pr.y - 0.5f * pr.w;
    float px2 = pr.x + 0.5f * pr.z, py2 = pr.y + 0.5f * pr.w;
    float w = fmaxf(fminf(gx2, px2) - fmaxf(gx1, px1), 0.f);
    float h = fmaxf(fminf(gy2, py2) - fmaxf(gy1, py1), 0.f);
    float inter = w * h;
    float ab = (px2 - px1) * (py2 - py1);
    float iou = inter / (aa + ab - inter);
    if (iou > best) { best = iou; bi = p; }   // ascending p -> first occurrence
  }
  sv[tid] = best; si[tid] = bi;
  __syncthreads();
  for (int s = 128; s > 0; s >>= 1) {
    if (tid < s) {
      if (sv[tid + s] > sv[tid] ||
          (sv[tid + s] == sv[tid] && si[tid + s] < si[tid])) {
        sv[tid] = sv[tid + s]; si[tid] = si[tid + s];
      }
    }
    __syncthreads();
  }
  if (tid == 0) { bpidx[blk] = si[0]; bpvalid[blk] = (sv[0] >= 0.2f) ? 1 : 0; }
}

__global__ void k_anyvalid(const int* __restrict__ bpvalid, int* __restrict__ anyvalid) {
  int b = threadIdx.x;
  if (b < NB) {
    int a = 0;
    for (int o = 0; o < NO; ++o) a |= bpvalid[b * NO + o];
    anyvalid[b] = a;
  }
}

// ------------------------------------------------------- fused main pass
__global__ void k_main(const float* __restrict__ loc, const float* __restrict__ conf,
                       const float* __restrict__ landm, const float* __restrict__ dpt,
                       const float* __restrict__ priors,
                       const float* __restrict__ boxes_gt, const float* __restrict__ kypts_gt,
                       const float* __restrict__ depths_gt, const int* __restrict__ labels_gt,
                       const int* __restrict__ bpidx, const int* __restrict__ bpvalid,
                       const int* __restrict__ anyvalid,
                       float* __restrict__ lossc, float* __restrict__ part,
                       int* __restrict__ npos) {
  __shared__ float sbox[NO * 4];
  __shared__ float skyp[NO * 10];
  __shared__ float sdep[NO * 2];
  __shared__ int   slab[NO];
  __shared__ int   sidx[NO];
  __shared__ int   sval[NO];
  __shared__ int   sany;
  __shared__ int   scnt;
  __shared__ float wred[8][4];

  int tid = threadIdx.x;
  int b   = blockIdx.y;
#if defined(HAVE_ASYNC_LDS)
  // stage GT tensors to LDS with async copies (per-lane global->LDS, ASYNCcnt)
  if (tid < NO * 4)       async_copy_b32(&sbox[tid], boxes_gt + b * NO * 4 + tid);
  /* NO*10 = 320 dwords */ async_copy_b32(&skyp[tid], kypts_gt + b * NO * 10 + tid);
  if (tid < NO * 10 - 256) async_copy_b32(&skyp[tid + 256], kypts_gt + b * NO * 10 + tid + 256);
  if (tid < NO * 2)       async_copy_b32(&sdep[tid], depths_gt + b * NO * 2 + tid);
#else
  if (tid < NO * 4) sbox[tid] = boxes_gt[b * NO * 4 + tid];
  if (tid >= 128 && tid < 128 + NO * 2) sdep[tid - 128] = depths_gt[b * NO * 2 + (tid - 128)];
  for (int i = tid; i < NO * 10; i += 256) skyp[i] = kypts_gt[b * NO * 10 + i];
#endif
  if (tid < NO) {
    slab[tid] = labels_gt[b * NO + tid];
    sidx[tid] = bpidx[b * NO + tid];
    sval[tid] = bpvalid[b * NO + tid];
  }
  if (tid == 0) { sany = anyvalid[b]; scnt = 0; }
#if defined(HAVE_ASYNC_LDS)
  wait_async0();
#endif
  __syncthreads();

  int p = blockIdx.x * 256 + tid;
  float4 pr = reinterpret_cast<const float4*>(priors)[p];
  float px1 = pr.x - 0.5f * pr.z, py1 = pr.y - 0.5f * pr.w;
  float px2 = pr.x + 0.5f * pr.z, py2 = pr.y + 0.5f * pr.w;
  float areab = (px2 - px1) * (py2 - py1);

  // best GT per prior (first occurrence on ties, like jnp.argmax)
  float best = -1.f; int bi = 0;
  for (int o = 0; o < NO; ++o) {
    float gx1 = sbox[o * 4 + 0], gy1 = sbox[o * 4 + 1];
    float gx2 = sbox[o * 4 + 2], gy2 = sbox[o * 4 + 3];
    float w = fmaxf(fminf(gx2, px2) - fmaxf(gx1, px1), 0.f);
    float h = fmaxf(fminf(gy2, py2) - fmaxf(gy1, py1), 0.f);
    float inter = w * h;
    float iou = inter / ((gx2 - gx1) * (gy2 - gy1) + areab - inter);
    if (iou > best) { best = iou; bi = o; }
  }
  // forced override: each valid GT keeps its best prior
  for (int o = 0; o < NO; ++o)
    if (sval[o] && sidx[o] == p) { best = 2.f; bi = o; }

  int cf = (best < THRESH) ? 0 : slab[bi];
  if (!sany) cf = 0;
  bool pos = cf > 0;

  long long base = (long long)b * NP + p;
  float c0 = conf[base * 2 + 0], c1 = conf[base * 2 + 1];
  float mx  = fmaxf(c0, c1);
  float lse = mx + logf(expf(c0 - mx) + expf(c1 - mx));
  float lc  = pos ? 0.f : (lse - c0);      // CE surrogate (== ce for negatives)
  lossc[base] = lc;
  float posce = pos ? (lse - c1) : 0.f;    // ce for positives (label = 1)

  float lloc = 0.f, llan = 0.f, ldep = 0.f;
  if (pos) {
    float gx1 = sbox[bi * 4 + 0], gy1 = sbox[bi * 4 + 1];
    float gx2 = sbox[bi * 4 + 2], gy2 = sbox[bi * 4 + 3];
    float tx = ((gx1 + gx2) * 0.5f - pr.x) / (VAR0 * pr.z);
    float ty = ((gy1 + gy2) * 0.5f - pr.y) / (VAR0 * pr.w);
    float tw = logf((gx2 - gx1) / pr.z) / VAR1;
    float th = logf((gy2 - gy1) / pr.w) / VAR1;
    float4 l4 = reinterpret_cast<const float4*>(loc)[base];
    lloc = smoothl1(l4.x - tx) + smoothl1(l4.y - ty) +
           smoothl1(l4.z - tw) + smoothl1(l4.w - th);
    for (int j = 0; j < 5; ++j) {
      float gx = (skyp[bi * 10 + 2 * j + 0] - pr.x) / (VAR0 * pr.z);
      float gy = (skyp[bi * 10 + 2 * j + 1] - pr.y) / (VAR0 * pr.w);
      llan += smoothl1(gx - landm[base * 10 + 2 * j + 0]);
      llan += smoothl1(gy - landm[base * 10 + 2 * j + 1]);
    }
    float d0 = sdep[bi * 2 + 0] - dpt[base * 2 + 0];
    float d1 = sdep[bi * 2 + 1] - dpt[base * 2 + 1];
    ldep = d0 * d0 + d1 * d1;
    atomicAdd(&scnt, 1);
  }

  // deterministic block reduction: WMMA per wave, then 8-wave LDS combine
  float r0 = wave_sum32(lloc);
  float r1 = wave_sum32(llan);
  float r2 = wave_sum32(ldep);
  float r3 = wave_sum32(posce);
  int w = tid >> 5, lane = tid & 31;
  if (lane == 0) { wred[w][0] = r0; wred[w][1] = r1; wred[w][2] = r2; wred[w][3] = r3; }
  __syncthreads();
  if (tid == 0) {
    float s0 = 0, s1 = 0, s2 = 0, s3 = 0;
    for (int i = 0; i < 8; ++i) { s0 += wred[i][0]; s1 += wred[i][1]; s2 += wred[i][2]; s3 += wred[i][3]; }
    int blk = blockIdx.x;
    part[0 * NB * 256 + b * 256 + blk] = s0;
    part[1 * NB * 256 + b * 256 + blk] = s1;
    part[2 * NB * 256 + b * 256 + blk] = s2;
    part[3 * NB * 256 + b * 256 + blk] = s3;
    atomicAdd(&npos[b], scnt);
  }
}

// -------------------------------------- radix-histogram selection (4x 8-bit passes)
__global__ void k_hist(const unsigned* __restrict__ keys, unsigned* __restrict__ hist,
                       const unsigned* __restrict__ prefix, const int* __restrict__ done,
                       int pass) {
  __shared__ unsigned lh[256];
  int tid = threadIdx.x, b = blockIdx.y;
  lh[tid] = 0u;
  __syncthreads();
  if (!done[b]) {
    int shift = 24 - 8 * pass;
    unsigned pfx = (pass == 0) ? 0u : prefix[b];
    long long base = (long long)b * NP + blockIdx.x * 1024 + tid;
    for (int i = 0; i < 4; ++i) {
      unsigned k = keys[base + i * 256];
      bool match = (pass == 0) || (((k ^ pfx) >> (shift + 8)) == 0u);
      if (match) atomicAdd(&lh[(k >> shift) & 255u], 1u);
    }
  }
  __syncthreads();
  if (lh[tid]) atomicAdd(&hist[b * 256 + tid], lh[tid]);
}

__global__ void k_scan(unsigned* __restrict__ hist, const int* __restrict__ npos,
                       int* __restrict__ target, unsigned* __restrict__ prefix,
                       int* __restrict__ done, int pass) {
  int b = threadIdx.x;
  if (b >= NB) return;
  if (pass == 0) {
    long long nn = (long long)NEGPOS * (long long)npos[b];
    int numneg = (nn > (NP - 1)) ? (NP - 1) : (int)nn;   // jnp.clip(a_max=P-1)
    if (numneg <= 0) { done[b] = 1; target[b] = 0; prefix[b] = 0xFFFFFFFFu; }
    else             { target[b] = numneg; prefix[b] = 0u; }
  }
  unsigned* h = hist + b * 256;
  if (!done[b]) {
    int shift = 24 - 8 * pass;
    int tgt = target[b];
    int acc = 0, chosen = 0;
    for (int v = 255; v >= 0; --v) {
      int c = (int)h[v];
      if (acc + c >= tgt) { chosen = v; break; }
      acc += c;
    }
    prefix[b] |= ((unsigned)chosen) << shift;
    target[b] = tgt - acc;          // after pass 3: # of tied keys to include
  }
  for (int v = 0; v < 256; ++v) h[v] = 0u;   // reset for next pass
}

// -------------------------------------------- sum of strictly-above-threshold negatives
__global__ void k_negsum(const float* __restrict__ lossc, const unsigned* __restrict__ prefix,
                         const int* __restrict__ done, float* __restrict__ negpart) {
  __shared__ float wred[8];
  int tid = threadIdx.x, b = blockIdx.y;
  float s = 0.f;
  int dn = done[b];
  unsigned tk = prefix[b];
  long long base = (long long)b * NP + blockIdx.x * 1024 + tid;
  if (!dn) {
    for (int i = 0; i < 4; ++i) {
      float v = lossc[base + i * 256];
      if (__float_as_uint(v) > tk) s += v;   // bits compare == float compare (all >= 0)
    }
  }
  float r = wave_sum32(s);
  int w = tid >> 5, lane = tid & 31;
  if (lane == 0) wred[w] = r;
  __syncthreads();
  if (tid == 0) {
    float t = 0.f;
    for (int i = 0; i < 8; ++i) t += wred[i];
    negpart[b * 64 + blockIdx.x] = t;
  }
}

// ---------------------------------------------------------------- final combine
__global__ void k_final(const float* __restrict__ part, const float* __restrict__ negpart,
                        const int* __restrict__ npos, const int* __restrict__ target,
                        const unsigned* __restrict__ prefix, const int* __restrict__ done,
                        float* __restrict__ out) {
  __shared__ float res[8];
  int tid = threadIdx.x, w = tid >> 5, lane = tid & 31;
  float s = 0.f;
  if (w < 4) {                         // wave w reduces quantity w (16*256 partials)
    for (int i = lane; i < NB * 256; i += 32) s += part[w * NB * 256 + i];
  } else if (w == 4) {                 // wave 4 reduces the negative-CE partials
    for (int i = lane; i < NB * 64; i += 32) s += negpart[i];
  }
  float r = wave_sum32(s);
  if (lane == 0) res[w] = r;
  __syncthreads();
  if (tid == 0) {
    long long npt = 0;
    for (int b2 = 0; b2 < NB; ++b2) npt += npos[b2];
    float tie = 0.f;                   // tied negatives all contribute value t
    for (int b2 = 0; b2 < NB; ++b2)
      if (!done[b2]) tie += (float)target[b2] * __uint_as_float(prefix[b2]);
    float n  = fmaxf((float)npt, 1.f);
    float n1 = fmaxf((float)(npt * 10), 1.f);
    float n2 = fmaxf((float)(npt * 2), 1.f);
    out[0] = res[0] / n;               // loss_l / n
    out[1] = (res[3] + res[4] + tie) / n;   // loss_cls / n
    out[2] = res[1] / n1;              // loss_landm / n1
    out[3] = res[2] / n2;              // loss_dpth / ndpth
  }
}

// ---------------------------------------------------------------- launch
extern "C" void kernel_launch(void* const* d_in, const int* in_sizes, int n_in,
                              void* d_out, int out_size, void* d_ws, size_t ws_size,
                              hipStream_t stream) {
  const float* loc    = (const float*)d_in[0];
  const float* conf   = (const float*)d_in[1];
  const float* landm  = (const float*)d_in[2];
  const float* dpt    = (const float*)d_in[3];
  const float* priors = (const float*)d_in[4];
  const float* boxes  = (const float*)d_in[5];
  const float* kypts  = (const float*)d_in[6];
  const float* depths = (const float*)d_in[7];
  const int*   labels = (const int*)d_in[8];
  float* out = (float*)d_out;

  // workspace carve-up (~4.3 MB total)
  float*    lossc    = (float*)d_ws;                 // NB*NP
  float*    part     = lossc + NB * NP;              // 4*NB*256
  float*    negpart  = part + 4 * NB * 256;          // NB*64
  unsigned* hist     = (unsigned*)(negpart + NB * 64); // NB*256
  int*      npos     = (int*)(hist + NB * 256);      // NB
  int*      target   = npos + NB;                    // NB
  unsigned* prefix   = (unsigned*)(target + NB);     // NB
  int*      done     = (int*)(prefix + NB);          // NB
  int*      bpidx    = done + NB;                    // NB*NO
  int*      bpvalid  = bpidx + NB * NO;              // NB*NO
  int*      anyvalid = bpvalid + NB * NO;            // NB

  k_init<<<1, 256, 0, stream>>>(hist, npos, done);
  k_bestprior<<<NB * NO, 256, 0, stream>>>(priors, boxes, bpidx, bpvalid);
  k_anyvalid<<<1, 32, 0, stream>>>(bpvalid, anyvalid);
  k_main<<<dim3(NP / 256, NB), 256, 0, stream>>>(
      loc, conf, landm, dpt, priors, boxes, kypts, depths, labels,
      bpidx, bpvalid, anyvalid, lossc, part, npos);
  for (int pass = 0; pass < 4; ++pass) {
    k_hist<<<dim3(NP / 1024, NB), 256, 0, stream>>>(
        (const unsigned*)lossc, hist, prefix, done, pass);
    k_scan<<<1, 32, 0, stream>>>(hist, npos, target, prefix, done, pass);
  }
  k_negsum<<<dim3(NP / 1024, NB), 256, 0, stream>>>(lossc, prefix, done, negpart);
  k_final<<<1, 256, 0, stream>>>(part, negpart, npos, target, prefix, done, out);
}